// GraphConvoluationSparse_62551903699211
// MI455X (gfx1250) — compile-verified
//
#include <hip/hip_runtime.h>
#include <hip/hip_bf16.h>

typedef float v2f __attribute__((ext_vector_type(2)));
typedef float v8f __attribute__((ext_vector_type(8)));

#define IN_DIM   256
#define OUT_DIM  256
#define LDSTRIDE 264   // 264 % 32 == 8 -> half-wave bank ranges disjoint for row,row+2 loads

// ---------------------------------------------------------------------------
// Kernel 1: hidden = x @ W, fp32 WMMA 16x16x4.
// Block = 256 threads = 8 waves. Wave w owns M-tile (blockIdx.x*8 + w): 16 rows,
// all 256 output columns (16 N-tiles, 16 x v8f accumulators = 128 VGPRs).
// Full W (256x256) staged in LDS with padded stride; x read exactly once.
// ---------------------------------------------------------------------------
__global__ __launch_bounds__(256)
void gcn_gemm_wmma(const float* __restrict__ x, const float* __restrict__ W,
                   float* __restrict__ hidden, int mtiles) {
    __shared__ float ldsW[IN_DIM * LDSTRIDE];   // 270,336 bytes (<320KB WGP LDS)

    const int t = threadIdx.x;

    // Stage W into LDS, float4 per iteration (64 float4 per row).
    const float4* W4 = (const float4*)W;
    #pragma unroll 4
    for (int i = t; i < IN_DIM * (OUT_DIM / 4); i += 256) {
        int r  = i >> 6;        // row (K index)
        int c4 = i & 63;        // float4 column
        float4 w = W4[i];
        *(float4*)&ldsW[r * LDSTRIDE + c4 * 4] = w;
    }
    __syncthreads();

    const int wave  = t >> 5;
    const int lane  = t & 31;
    const int mtile = blockIdx.x * 8 + wave;
    if (mtile >= mtiles) return;

    const int n  = lane & 15;   // N column within tile; also M row for A frag
    const int hi = lane >> 4;   // half-wave select

    const float* arow = x + (size_t)(mtile * 16 + n) * IN_DIM;

    const v8f vzero = {0.f, 0.f, 0.f, 0.f, 0.f, 0.f, 0.f, 0.f};
    v8f acc[16];
    #pragma unroll
    for (int i = 0; i < 16; ++i) acc[i] = vzero;

    for (int k = 0; k < IN_DIM; k += 4) {
        // A fragment: lane(m=n): VGPR0 = A[m][k+2*hi], VGPR1 = A[m][k+2*hi+1]
        v2f a = *(const v2f*)(arow + k + 2 * hi);

        const float* brow = &ldsW[(k + 2 * hi) * LDSTRIDE + n];
        #pragma unroll
        for (int nt = 0; nt < 16; ++nt) {
            // B fragment: VGPR0 = W[k+2*hi][nt*16+n], VGPR1 = W[k+2*hi+1][nt*16+n]
            v2f b;
            b.x = brow[nt * 16];
            b.y = brow[LDSTRIDE + nt * 16];
            acc[nt] = __builtin_amdgcn_wmma_f32_16x16x4_f32(
                false, a, false, b, (short)0, acc[nt], false, false);
        }
    }

    // C/D layout: VGPR r -> row (r + 8*hi), col n (within N-tile nt).
    float* orow = hidden + (size_t)(mtile * 16 + 8 * hi) * OUT_DIM + n;
    #pragma unroll
    for (int nt = 0; nt < 16; ++nt) {
        #pragma unroll
        for (int r = 0; r < 8; ++r) {
            orow[(size_t)r * OUT_DIM + nt * 16] = acc[nt][r];
        }
    }
}

// ---------------------------------------------------------------------------
// Kernel 2: out[i][:] = b[:]   (float4-vectorized bias broadcast)
// ---------------------------------------------------------------------------
__global__ __launch_bounds__(256)
void gcn_bias_init(const float4* __restrict__ b4, float4* __restrict__ out4,
                   int total4) {
    int i = blockIdx.x * 256 + threadIdx.x;
    if (i < total4) out4[i] = b4[i & (OUT_DIM / 4 - 1)];
}

// ---------------------------------------------------------------------------
// Kernel 3: COO SpMM scatter. One wave (32 lanes) per edge; each lane moves
// two float4s (2*32*16B = 256 floats), scaled by edge weight, scattered with
// fp32 global atomics (no return -> STOREcnt path).
// ---------------------------------------------------------------------------
__global__ __launch_bounds__(256)
void gcn_edge_scatter(const float* __restrict__ hidden,
                      const int* __restrict__ rows,
                      const int* __restrict__ cols,
                      const float* __restrict__ vals,
                      float* __restrict__ out, int nEdges) {
    const int wave = threadIdx.x >> 5;
    const int lane = threadIdx.x & 31;
    const int e = blockIdx.x * 8 + wave;
    if (e >= nEdges) return;

    const int   row = rows[e];
    const int   col = cols[e];
    const float v   = vals[e];

    const float4* src = (const float4*)(hidden + (size_t)col * OUT_DIM);
    float*        dst = out + (size_t)row * OUT_DIM;

    #pragma unroll
    for (int p = 0; p < 2; ++p) {
        float4 h = src[p * 32 + lane];
        int base = (p * 32 + lane) * 4;
        atomicAdd(dst + base + 0, v * h.x);
        atomicAdd(dst + base + 1, v * h.y);
        atomicAdd(dst + base + 2, v * h.z);
        atomicAdd(dst + base + 3, v * h.w);
    }
}

extern "C" void kernel_launch(void* const* d_in, const int* in_sizes, int n_in,
                              void* d_out, int out_size, void* d_ws, size_t ws_size,
                              hipStream_t stream) {
    const float* x    = (const float*)d_in[0];
    const int*   rows = (const int*)d_in[1];
    const int*   cols = (const int*)d_in[2];
    const float* vals = (const float*)d_in[3];
    const float* W    = (const float*)d_in[4];
    const float* b    = (const float*)d_in[5];
    float*       out  = (float*)d_out;
    float*       hidden = (float*)d_ws;          // N*256 floats = 51.2 MB scratch

    const int N = in_sizes[0] / IN_DIM;          // 50000
    const int E = in_sizes[1];                   // 800000

    // 1) hidden = x @ W   (WMMA fp32)
    const int mtiles = (N + 15) / 16;            // 3125
    const int gemmBlocks = (mtiles + 7) / 8;     // 391
    gcn_gemm_wmma<<<gemmBlocks, 256, 0, stream>>>(x, W, hidden, mtiles);

    // 2) out = broadcast(b)
    const int total4 = N * (OUT_DIM / 4);        // 3.2M float4
    gcn_bias_init<<<(total4 + 255) / 256, 256, 0, stream>>>(
        (const float4*)b, (float4*)out, total4);

    // 3) out[row] += val * hidden[col]  (atomic scatter)
    gcn_edge_scatter<<<(E + 7) / 8, 256, 0, stream>>>(
        hidden, rows, cols, vals, out, E);
}